// RefiedKBQA_72009421685057
// MI455X (gfx1250) — compile-verified
//
#include <hip/hip_runtime.h>
#include <hip/hip_bf16.h>

#define B      32
#define N_E    500000
#define N_T    2000000
#define N_R    200
#define N_W2V  300

typedef __attribute__((ext_vector_type(2))) float v2f;
typedef __attribute__((ext_vector_type(8))) float v8f;

// ---------------------------------------------------------------------------
// Transpose x [B, N_E] -> xT [N_E, B].  32x32 tiles via LDS, fully coalesced.
// ---------------------------------------------------------------------------
__global__ __launch_bounds__(256) void transpose_in_kernel(
    const float* __restrict__ x, float* __restrict__ xT) {
  __shared__ float tile[32][33];
  const int e0  = blockIdx.x * 32;
  const int col = threadIdx.x & 31;
  const int row = threadIdx.x >> 5;  // 0..7
#pragma unroll
  for (int k = 0; k < 4; ++k) {
    const int b = row + k * 8;
    tile[b][col] = x[(size_t)b * N_E + e0 + col];
  }
  __syncthreads();
#pragma unroll
  for (int k = 0; k < 4; ++k) {
    const int e = row + k * 8;
    xT[(size_t)(e0 + e) * 32 + col] = tile[col][e];
  }
}

// ---------------------------------------------------------------------------
// Transpose xT [N_E, B] -> out [B, N_E].
// ---------------------------------------------------------------------------
__global__ __launch_bounds__(256) void transpose_out_kernel(
    const float* __restrict__ xT, float* __restrict__ out) {
  __shared__ float tile[32][33];
  const int e0  = blockIdx.x * 32;
  const int col = threadIdx.x & 31;
  const int row = threadIdx.x >> 5;
#pragma unroll
  for (int k = 0; k < 4; ++k) {
    const int e = row + k * 8;
    tile[e][col] = xT[(size_t)(e0 + e) * 32 + col];   // col = batch
  }
  __syncthreads();
#pragma unroll
  for (int k = 0; k < 4; ++k) {
    const int b = row + k * 8;
    out[(size_t)b * N_E + e0 + col] = tile[col][b];   // col = entity
  }
}

// ---------------------------------------------------------------------------
// R_T[h][n, m] = (q @ Wh + bh)[m, n], stored relation-major [N_R, 32].
// One wave per 16x16 tile, V_WMMA_F32_16X16X4_F32 over K=300 (75 steps of 4).
// grid = 3 hops * 2 mtiles * 13 ntiles = 78 waves.
// Tail-tile handling is branch-free: loads use a clamped in-bounds column and
// the loaded value is selected to zero (v_cndmask), so the inner loop is a
// straight load/load/wmma stream with EXEC permanently all-1s.
// ---------------------------------------------------------------------------
__global__ __launch_bounds__(32) void qw_gemm_kernel(
    const float* __restrict__ q,
    const float* __restrict__ W1, const float* __restrict__ b1,
    const float* __restrict__ W2, const float* __restrict__ b2,
    const float* __restrict__ W3, const float* __restrict__ b3,
    float* __restrict__ R_T) {
  const int tile = blockIdx.x;
  const int h    = tile / 26;
  const int rem  = tile % 26;
  const int mt   = rem / 13;
  const int nt   = rem % 13;
  const float* W    = (h == 0) ? W1 : (h == 1) ? W2 : W3;
  const float* bias = (h == 0) ? b1 : (h == 1) ? b2 : b3;

  const int lane   = threadIdx.x;
  const int row    = mt * 16 + (lane & 15);     // batch row (M), 0..31
  const int col    = nt * 16 + (lane & 15);     // relation col (N)
  const bool colOK = (col < N_R);
  const int colc   = colOK ? col : (N_R - 1);   // clamped, always in-bounds
  const int khalf  = (lane >> 4) * 2;           // K sub-offset per half-wave

  v8f acc = {};
  for (int k = 0; k < N_W2V; k += 4) {
    v2f a, bf;
    a.x = q[(size_t)row * N_W2V + k + khalf];
    a.y = q[(size_t)row * N_W2V + k + khalf + 1];
    float w0 = W[(size_t)(k + khalf)     * N_R + colc];
    float w1 = W[(size_t)(k + khalf + 1) * N_R + colc];
    bf.x = colOK ? w0 : 0.0f;                   // branch-free select
    bf.y = colOK ? w1 : 0.0f;
    acc = __builtin_amdgcn_wmma_f32_16x16x4_f32(
        /*neg_a=*/false, a, /*neg_b=*/false, bf,
        /*c_mod=*/(short)0, acc, /*reuse_a=*/false, /*reuse_b=*/false);
  }

  if (colOK) {
    const float bv = bias[col];
#pragma unroll
    for (int i = 0; i < 8; ++i) {
      const int m = mt * 16 + i + (lane >> 4) * 8;   // C/D layout: M = i (+8 hi half)
      R_T[(size_t)h * N_R * 32 + (size_t)col * 32 + m] = acc[i] + bv;
    }
  }
}

// ---------------------------------------------------------------------------
// prep: xnext = (n_hop >= h) ? 0 : xcur  (elementwise, float4 vectorized)
// ---------------------------------------------------------------------------
__global__ __launch_bounds__(256) void prep_kernel(
    float4* __restrict__ xnext, const float4* __restrict__ xcur,
    const int* __restrict__ n_hop, int h) {
  const int i = blockIdx.x * 256 + threadIdx.x;   // N_E*32/4 = 4,000,000 elems
  if (i < N_E * 8) {
    if (*n_hop >= h) {
      xnext[i] = make_float4(0.f, 0.f, 0.f, 0.f);
    } else {
      xnext[i] = xcur[i];
    }
  }
}

// ---------------------------------------------------------------------------
// Hop scatter: lane b handles batch index b.  xT is entity-major so each
// triple's gather and atomic scatter are single 128B cache-line transactions.
// Relation matrix (relation-major, 25.6KB) lives in LDS.
// ---------------------------------------------------------------------------
#define WAVES_PER_BLOCK  8
#define TRIPLES_PER_WAVE 256

__global__ __launch_bounds__(256) void hop_scatter_kernel(
    const float* __restrict__ xT_cur, float* __restrict__ xT_next,
    const int* __restrict__ subj, const int* __restrict__ rel,
    const int* __restrict__ obj, const float* __restrict__ R_T_h,
    const int* __restrict__ n_hop, int h) {
  if (*n_hop < h) return;

  __shared__ float ldsR[N_R * 32];
  for (int i = threadIdx.x; i < N_R * 32; i += 256) ldsR[i] = R_T_h[i];
  __syncthreads();

  const int lane = threadIdx.x & 31;
  const int wave = blockIdx.x * WAVES_PER_BLOCK + (threadIdx.x >> 5);
  const int t0   = wave * TRIPLES_PER_WAVE;

  for (int c = 0; c < TRIPLES_PER_WAVE; c += 32) {
    const int base   = t0 + c;
    const int nvalid = min(32, N_T - base);
    if (nvalid <= 0) return;

    // Coalesced index loads: one triple per lane.
    int sv = 0, rv = 0, ov = 0;
    if (lane < nvalid) {
      const int t = base + lane;
      sv = subj[t];
      rv = rel[t];
      ov = obj[t];
    }

    for (int j = 0; j < nvalid; ++j) {
      const int s = __shfl(sv, j, 32);
      const int r = __shfl(rv, j, 32);
      const int o = __shfl(ov, j, 32);
      const float m = xT_cur[(size_t)s * 32 + lane] * ldsR[r * 32 + lane];
      atomicAdd(&xT_next[(size_t)o * 32 + lane], m);
    }
  }
}

// ---------------------------------------------------------------------------
extern "C" void kernel_launch(void* const* d_in, const int* in_sizes, int n_in,
                              void* d_out, int out_size, void* d_ws, size_t ws_size,
                              hipStream_t stream) {
  const float* x     = (const float*)d_in[0];
  const float* q     = (const float*)d_in[1];
  const int*   subj  = (const int*)d_in[2];
  const int*   rel   = (const int*)d_in[3];
  const int*   obj   = (const int*)d_in[4];
  const float* W1    = (const float*)d_in[5];
  const float* b1    = (const float*)d_in[6];
  const float* W2    = (const float*)d_in[7];
  const float* b2    = (const float*)d_in[8];
  const float* W3    = (const float*)d_in[9];
  const float* b3    = (const float*)d_in[10];
  const int*   n_hop = (const int*)d_in[11];
  float*       out   = (float*)d_out;

  float* xT_a = (float*)d_ws;                      // [N_E, 32]  64 MB
  float* xT_b = xT_a + (size_t)N_E * 32;           // [N_E, 32]  64 MB
  float* R_T  = xT_b + (size_t)N_E * 32;           // [3, N_R, 32]  76.8 KB

  const int TGRID = N_E / 32;                                    // 15625
  const int SGRID = (N_T + WAVES_PER_BLOCK * TRIPLES_PER_WAVE - 1) /
                    (WAVES_PER_BLOCK * TRIPLES_PER_WAVE);        // 977
  const int PGRID = (N_E * 8 + 255) / 256;                       // 15625

  // 1) state -> entity-major layout
  transpose_in_kernel<<<TGRID, 256, 0, stream>>>(x, xT_a);

  // 2) r_h = q @ W_h + b_h via WMMA f32 16x16x4, stored relation-major
  qw_gemm_kernel<<<78, 32, 0, stream>>>(q, W1, b1, W2, b2, W3, b3, R_T);

  // 3) hop 1: xT_a -> xT_b
  prep_kernel<<<PGRID, 256, 0, stream>>>((float4*)xT_b, (const float4*)xT_a, n_hop, 1);
  hop_scatter_kernel<<<SGRID, 256, 0, stream>>>(xT_a, xT_b, subj, rel, obj,
                                                R_T + (size_t)0 * N_R * 32, n_hop, 1);
  // 4) hop 2: xT_b -> xT_a
  prep_kernel<<<PGRID, 256, 0, stream>>>((float4*)xT_a, (const float4*)xT_b, n_hop, 2);
  hop_scatter_kernel<<<SGRID, 256, 0, stream>>>(xT_b, xT_a, subj, rel, obj,
                                                R_T + (size_t)1 * N_R * 32, n_hop, 2);
  // 5) hop 3: xT_a -> xT_b
  prep_kernel<<<PGRID, 256, 0, stream>>>((float4*)xT_b, (const float4*)xT_a, n_hop, 3);
  hop_scatter_kernel<<<SGRID, 256, 0, stream>>>(xT_a, xT_b, subj, rel, obj,
                                                R_T + (size_t)2 * N_R * 32, n_hop, 3);

  // 6) back to [B, N_E]
  transpose_out_kernel<<<TGRID, 256, 0, stream>>>(xT_b, out);
}